// GatedTCNStack_33225867002564
// MI455X (gfx1250) — compile-verified
//
#include <hip/hip_runtime.h>
#include <math.h>

typedef __attribute__((ext_vector_type(2))) float v2f;
typedef __attribute__((ext_vector_type(8))) float v8f;

#define NB    64
#define C32   32
#define TT    13
#define NN    512
#define CSK   256
#define NLAY  8
#define PP    208           // TT * 16 columns per block tile
#define PR    416           // pair-row stride = 2*PP (K-pair interleaved layouts)

// ---- LDS layout (float offsets) ----
// xbuf : [16 cpair][15 tp][16 nl][2]  pair-interleaved x tile (2 leading zero t-slots)
// gates: [128 kpair][208 j][2]        pair-interleaved gates, kk = l*32 + c
// yb   : [16 mpair][208 j][2]         conv-b pre-activation scratch
#define XBUF_OFF   0        // 16*15*32          = 7680
#define GATE_OFF   7680     // 128*416           = 53248
#define YB_OFF     60928    // 16*416            = 6656
#define BSUM_OFF   67584    // 256  sum over layers of bs
#define BA_OFF     67840    // 256  ba flat
#define BB_OFF     68096    // 256  bb flat
#define LDS_FLOATS 68352    // 273,408 bytes (<= 320KB WGP LDS)

// x tile address: channel c, padded time tp, lane column nl
__device__ __forceinline__ int x2(int c, int tp, int nl) {
  return XBUF_OFF + (c >> 1) * 480 + tp * 32 + (nl << 1) + (c & 1);
}

__device__ __forceinline__ v8f wmma4(v2f a, v2f b, v8f c) {
  return __builtin_amdgcn_wmma_f32_16x16x4_f32(false, a, false, b, (short)0, c, false, false);
}

// Dilated causal conv tile job: y(16 x NT*16) = [Wk0|Wk1](16x64) @ [x(t-d);x(t)](64x...)
template<int NT>
__device__ __forceinline__ void conv_do(float* smem, const float* __restrict__ W,
                                        int dstPairBase, int d, int mt, int lo, int hi,
                                        int ntbase)
{
  const v8f vzero = {0.f, 0.f, 0.f, 0.f, 0.f, 0.f, 0.f, 0.f};
  v8f acc[NT];
  const int om = mt * 16 + lo;

  // ---- peeled ks = 0: C folds to WMMA's inline-0 operand ----
  {
    const int k0 = hi * 2;               // tap 0, c = k0 (even)
    v2f av;
    av.x = W[(om * C32 + k0) * 2];
    av.y = W[(om * C32 + k0 + 1) * 2];
#pragma unroll
    for (int i = 0; i < NT; ++i) {
      const int tp = ntbase + i + 2 - d;
      const float2 xv = *reinterpret_cast<const float2*>(
          &smem[XBUF_OFF + (k0 >> 1) * 480 + tp * 32 + lo * 2]);
      v2f bv; bv.x = xv.x; bv.y = xv.y;
      acc[i] = wmma4(av, bv, vzero);
    }
  }
#pragma unroll 1
  for (int ks = 1; ks < 16; ++ks) {      // K = 64 = CIN * Ktaps, rolled (reg pressure)
    const int k0  = ks * 4 + hi * 2;     // even: (c0, c0+1) share the tap
    const int c0  = k0 & 31;
    const int tap = k0 >> 5;
    v2f av;
    av.x = W[(om * C32 + c0) * 2 + tap];
    av.y = W[(om * C32 + c0 + 1) * 2 + tap];
#pragma unroll
    for (int i = 0; i < NT; ++i) {       // NT independent accumulator chains hide latency
      const int tp = ntbase + i + 2 - d * (1 - tap);
      const float2 xv = *reinterpret_cast<const float2*>(
          &smem[XBUF_OFF + (c0 >> 1) * 480 + tp * 32 + lo * 2]);   // one ds_load_b64
      v2f bv; bv.x = xv.x; bv.y = xv.y;
      acc[i] = wmma4(av, bv, acc[i]);
    }
  }
#pragma unroll
  for (int i = 0; i < NT; ++i) {
    const int j2 = ((ntbase + i) * 16 + lo) * 2;
#pragma unroll
    for (int r = 0; r < 8; r += 2) {     // C/D rows pair up (m even) -> float2 stores
      const int m0 = mt * 16 + r + 8 * hi;
      float2 s; s.x = acc[i][r]; s.y = acc[i][r + 1];
      *reinterpret_cast<float2*>(&smem[dstPairBase + (m0 >> 1) * PR + j2]) = s;
    }
  }
}

// Skip projection tile job: skip(16 x NT*16) = [Ws0..Ws7](16x256) @ gates(256x...)
template<int NT>
__device__ __forceinline__ void skip_do(const float* smem, const float* __restrict__ Ws,
                                        float* __restrict__ skip_out, size_t outBase,
                                        int mt, int lo, int hi, int ntbase)
{
  const v8f vzero = {0.f, 0.f, 0.f, 0.f, 0.f, 0.f, 0.f, 0.f};
  v8f acc[NT];
  const int om = mt * 16 + lo;

  // ---- peeled ks = 0 ----
  {
    const int k0 = hi * 2;
    const float2 w = *reinterpret_cast<const float2*>(Ws + (size_t)om * C32 + k0);
    v2f av; av.x = w.x; av.y = w.y;
#pragma unroll
    for (int i = 0; i < NT; ++i) {
      const int j = (ntbase + i) * 16 + lo;
      const float2 g2 = *reinterpret_cast<const float2*>(
          &smem[GATE_OFF + (k0 >> 1) * PR + j * 2]);
      v2f bv; bv.x = g2.x; bv.y = g2.y;
      acc[i] = wmma4(av, bv, vzero);
    }
  }
#pragma unroll 1
  for (int ks = 1; ks < 64; ++ks) {      // K = 256 = NLAY * C32
    const int k0 = ks * 4 + hi * 2;      // even -> contiguous pairs everywhere
    const float2 w = *reinterpret_cast<const float2*>(
        Ws + ((size_t)(k0 >> 5) * CSK + om) * C32 + (k0 & 31));    // one b64 A load
    v2f av; av.x = w.x; av.y = w.y;
#pragma unroll
    for (int i = 0; i < NT; ++i) {
      const int j = (ntbase + i) * 16 + lo;
      const float2 g2 = *reinterpret_cast<const float2*>(
          &smem[GATE_OFF + (k0 >> 1) * PR + j * 2]);               // one ds_load_b64
      v2f bv; bv.x = g2.x; bv.y = g2.y;
      acc[i] = wmma4(av, bv, acc[i]);
    }
  }
#pragma unroll
  for (int i = 0; i < NT; ++i) {
    const int t = ntbase + i;
#pragma unroll
    for (int r = 0; r < 8; ++r) {
      const int m = mt * 16 + r + 8 * hi;
      skip_out[outBase + ((size_t)m * TT + t) * NN + lo] = acc[i][r] + smem[BSUM_OFF + m];
    }
  }
}

extern "C" __global__ void __launch_bounds__(256, 1)
gated_tcn_wmma(const float* __restrict__ xin,
               const float* __restrict__ Wa, const float* __restrict__ ba,
               const float* __restrict__ Wb, const float* __restrict__ bb,
               const float* __restrict__ Ws, const float* __restrict__ bs,
               float* __restrict__ skip_out, float* __restrict__ x_out)
{
  extern __shared__ float smem[];
  const int tid  = threadIdx.x;
  const int lane = tid & 31;
  // force wave id into an SGPR so all job selection is scalar control flow
  const int wave = __builtin_amdgcn_readfirstlane(tid >> 5);
  const int lo   = lane & 15;
  const int hi   = lane >> 4;
  const int b    = blockIdx.x >> 5;
  const int n0   = (blockIdx.x & 31) << 4;

  // ---- stage biases in LDS: Σ_l bs  |  ba flat  |  bb flat ----
  {
    float s = 0.f;
#pragma unroll
    for (int l = 0; l < NLAY; ++l) s += bs[l * CSK + tid];
    smem[BSUM_OFF + tid] = s;
    smem[BA_OFF + tid] = ba[tid];   // tid spans exactly NLAY*C32 = 256
    smem[BB_OFF + tid] = bb[tid];
  }

  // ---- zero causal pad slots (tp = 0,1), float2 per job ----
  for (int idx = tid; idx < 512; idx += 256) {
    const int cp = idx >> 5;
    const int r  = idx & 31;            // r = tp*16 + nl
    float2 z; z.x = 0.f; z.y = 0.f;
    *reinterpret_cast<float2*>(&smem[XBUF_OFF + cp * 480 + (r >> 4) * 32 + (r & 15) * 2]) = z;
  }

  // ---- load x tile: two channels' float4 -> four interleaved float2 LDS stores ----
  for (int idx = tid; idx < 16 * TT * 4; idx += 256) {
    const int q  = idx & 3;
    const int t  = (idx >> 2) % TT;
    const int cp = idx / (TT * 4);
    const float4 v0 = *reinterpret_cast<const float4*>(
        xin + (((size_t)b * C32 + cp * 2) * TT + t) * NN + n0 + q * 4);
    const float4 v1 = *reinterpret_cast<const float4*>(
        xin + (((size_t)b * C32 + cp * 2 + 1) * TT + t) * NN + n0 + q * 4);
    const int base = XBUF_OFF + cp * 480 + (t + 2) * 32 + q * 8;
    float2 s;
    s.x = v0.x; s.y = v1.x; *reinterpret_cast<float2*>(&smem[base + 0]) = s;
    s.x = v0.y; s.y = v1.y; *reinterpret_cast<float2*>(&smem[base + 2]) = s;
    s.x = v0.z; s.y = v1.z; *reinterpret_cast<float2*>(&smem[base + 4]) = s;
    s.x = v0.w; s.y = v1.w; *reinterpret_cast<float2*>(&smem[base + 6]) = s;
  }
  __syncthreads();

  // ==================== 8-layer stack, fully in LDS ====================
#pragma unroll 1
  for (int l = 0; l < NLAY; ++l) {
    const int d = (l & 1) ? 2 : 1;

    // 4 (conv, mtile) combos over 8 waves; wave pairs split 13 n-tiles as 7 + 6
    const int  combo = wave >> 1;     // 0: ya mt0, 1: ya mt1, 2: yb mt0, 3: yb mt1
    const int  par   = wave & 1;
    const bool isB   = combo >= 2;
    const int  mt    = combo & 1;
    const float* W   = (isB ? Wb : Wa) + l * C32 * C32 * 2;   // [o][c][k]
    const int dstPairBase = isB ? YB_OFF : (GATE_OFF + l * 16 * PR);

    if (par == 0) conv_do<7>(smem, W, dstPairBase, d, mt, lo, hi, 0);
    else          conv_do<6>(smem, W, dstPairBase, d, mt, lo, hi, 7);
    __syncthreads();

    // gate = tanh(ya+ba) * sigmoid(yb+bb); residual x += gate  (channel-pair VALU)
    for (int e = tid; e < 16 * PP; e += 256) {
      const int cp = e / PP;            // channel pair 0..15
      const int j  = e - cp * PP;
      const int gidx = GATE_OFF + (l * 16 + cp) * PR + j * 2;
      const int yidx = YB_OFF + cp * PR + j * 2;
      const float2 ya2 = *reinterpret_cast<const float2*>(&smem[gidx]);
      const float2 yb2 = *reinterpret_cast<const float2*>(&smem[yidx]);
      const float2 bA  = *reinterpret_cast<const float2*>(&smem[BA_OFF + l * C32 + cp * 2]);
      const float2 bB  = *reinterpret_cast<const float2*>(&smem[BB_OFF + l * C32 + cp * 2]);
#if __has_builtin(__builtin_amdgcn_tanhf)
      const float t0 = __builtin_amdgcn_tanhf(ya2.x + bA.x);
      const float t1 = __builtin_amdgcn_tanhf(ya2.y + bA.y);
#else
      const float t0 = tanhf(ya2.x + bA.x);
      const float t1 = tanhf(ya2.y + bA.y);
#endif
      const float g0 = t0 * (1.f / (1.f + __expf(-(yb2.x + bB.x))));
      const float g1 = t1 * (1.f / (1.f + __expf(-(yb2.y + bB.y))));
      float2 g2; g2.x = g0; g2.y = g1;
      *reinterpret_cast<float2*>(&smem[gidx]) = g2;        // keep gate for skip GEMM
      const int xidx = XBUF_OFF + cp * 480 + (j >> 4) * 32 + 64 + (j & 15) * 2;
      float2 xv = *reinterpret_cast<const float2*>(&smem[xidx]);
      xv.x += g0; xv.y += g1;
      *reinterpret_cast<float2*>(&smem[xidx]) = xv;        // residual stream
    }
    __syncthreads();
  }

  // ---- write final residual stream x (de-interleave back to [c][t][n]) ----
  for (int idx = tid; idx < 16 * TT * 4; idx += 256) {
    const int q  = idx & 3;
    const int t  = (idx >> 2) % TT;
    const int cp = idx / (TT * 4);
    const int base = XBUF_OFF + cp * 480 + (t + 2) * 32 + q * 8;
    const float2 p0 = *reinterpret_cast<const float2*>(&smem[base + 0]);
    const float2 p1 = *reinterpret_cast<const float2*>(&smem[base + 2]);
    const float2 p2 = *reinterpret_cast<const float2*>(&smem[base + 4]);
    const float2 p3 = *reinterpret_cast<const float2*>(&smem[base + 6]);
    float4 v0; v0.x = p0.x; v0.y = p1.x; v0.z = p2.x; v0.w = p3.x;
    float4 v1; v1.x = p0.y; v1.y = p1.y; v1.z = p2.y; v1.w = p3.y;
    *reinterpret_cast<float4*>(
        x_out + (((size_t)b * C32 + cp * 2) * TT + t) * NN + n0 + q * 4) = v0;
    *reinterpret_cast<float4*>(
        x_out + (((size_t)b * C32 + cp * 2 + 1) * TT + t) * NN + n0 + q * 4) = v1;
  }

  // ---- fused skip projection: skip(256x208) = [Ws0..Ws7](256x256) @ gates(256x208) ----
  const size_t outBase = (size_t)b * CSK * TT * NN + n0;
#pragma unroll 1
  for (int pass = 0; pass < 2; ++pass) {
    const int mt = wave + pass * 8;          // 16 M-tiles over 8 waves, 2 passes
    skip_do<7>(smem, Ws, skip_out, outBase, mt, lo, hi, 0);
    skip_do<6>(smem, Ws, skip_out, outBase, mt, lo, hi, 7);
  }
}

extern "C" void kernel_launch(void* const* d_in, const int* in_sizes, int n_in,
                              void* d_out, int out_size, void* d_ws, size_t ws_size,
                              hipStream_t stream) {
  (void)in_sizes; (void)n_in; (void)out_size; (void)d_ws; (void)ws_size;
  const float* x  = (const float*)d_in[0];
  const float* Wa = (const float*)d_in[1];
  const float* ba = (const float*)d_in[2];
  const float* Wb = (const float*)d_in[3];
  const float* bb = (const float*)d_in[4];
  const float* Ws = (const float*)d_in[5];
  const float* bs = (const float*)d_in[6];
  float* skip = (float*)d_out;
  float* xout = skip + (size_t)NB * CSK * TT * NN;

  const size_t lds_bytes = (size_t)LDS_FLOATS * sizeof(float);
  (void)hipFuncSetAttribute(reinterpret_cast<const void*>(gated_tcn_wmma),
                            hipFuncAttributeMaxDynamicSharedMemorySize,
                            (int)lds_bytes);

  dim3 grid(NB * (NN / 16));   // 2048 blocks: one (batch, 16-wide n slice) each
  dim3 blk(256);               // 8 wave32s
  gated_tcn_wmma<<<grid, blk, lds_bytes, stream>>>(x, Wa, ba, Wb, bb, Ws, bs, skip, xout);
}